// HyperGRUCell_28174985461972
// MI455X (gfx1250) — compile-verified
//
#include <hip/hip_runtime.h>

// HyperGRU cell: per-batch GEMVs vs. per-batch weights, bandwidth-bound
// (~1.61 GB streamed once, ~0.8 GFLOP => 0.5 FLOP/B; ideal ~69us @ 23.3 TB/s).
// Strategy: V_WMMA_F32_16X16X4_F32 with A (input vector chunk) replicated
// across M rows; each wave handles 32 output columns (two 16-col B fragments)
// so weight loads consume full 128B cachelines promptly.

typedef __attribute__((ext_vector_type(2))) float v2f;
typedef __attribute__((ext_vector_type(8))) float v8f;

#define UNITS     512
#define INPUT_DIM 512
#define BATCH     256
#define THREE_U   (3 * UNITS)              // 1536
#define NUM_K     (INPUT_DIM * THREE_U)    // 786432
#define NUM_RK    (UNITS * THREE_U)        // 786432
#define PARAM_DIM (NUM_K + NUM_RK + 2 * THREE_U)

__device__ __forceinline__ float hsig(float v) {
    return fminf(fmaxf(fmaf(v, 0.2f, 0.5f), 0.0f), 1.0f);
}

__global__ __launch_bounds__(32) void hypergru_wmma(
    const float* __restrict__ inputs,
    const float* __restrict__ params,
    const float* __restrict__ h_tm1,
    float* __restrict__ out)
{
    __shared__ float sx[INPUT_DIM];
    __shared__ float sh[UNITS];

    const int lane = threadIdx.x;        // 0..31
    const int b    = blockIdx.x >> 4;    // 256 batches
    const int ut   = blockIdx.x & 15;    // 16 tiles of 32 units

    // Stage the two input vectors into LDS (broadcast source for A fragments).
    const float* xg = inputs + (size_t)b * INPUT_DIM;
    const float* hg = h_tm1 + (size_t)b * UNITS;
#pragma unroll
    for (int i = 0; i < INPUT_DIM; i += 32) {
        sx[i + lane] = xg[i + lane];
        sh[i + lane] = hg[i + lane];
    }
    __syncthreads();

    const float* Wb = params + (size_t)b * PARAM_DIM;
    const int hh16 = lane >> 4;   // half-wave: selects K-row pairing
    const int cl   = lane & 15;   // column within a 16-col fragment

    // 6 passes: p=0..2 -> kernel (x) gates z,r,h ; p=3..5 -> recurrent (h) gates.
    float res[6];
#pragma unroll
    for (int p = 0; p < 6; ++p) {
        const float* W  = Wb + (p < 3 ? 0 : NUM_K) + (size_t)(p % 3) * UNITS + ut * 32;
        const float* sv = (p < 3) ? sx : sh;

        // B fragment guess (mirror of documented A layout):
        //   VGPR0: lanes0-15 K=0, lanes16-31 K=2 ; VGPR1: K=1 / K=3, N = lane&15.
        // We load rows {i0+h, i0+2+h} and build A with the matching K->row map,
        // so the contraction sums x[i0+r]*W[i0+r, n] for r=0..3.
        const float* pW = W + hh16 * THREE_U + cl;
        const float* pA = sv + hh16;

        v8f accL = {0.f, 0.f, 0.f, 0.f, 0.f, 0.f, 0.f, 0.f};
        v8f accR = {0.f, 0.f, 0.f, 0.f, 0.f, 0.f, 0.f, 0.f};

#pragma unroll 4
        for (int i0 = 0; i0 < INPUT_DIM; i0 += 4) {
            v2f A, BL, BR;
            BL.x = pW[0];                   // row i0+h,   cols [t,   t+15]
            BL.y = pW[2 * THREE_U];         // row i0+2+h
            BR.x = pW[16];                  // row i0+h,   cols [t+16,t+31]
            BR.y = pW[2 * THREE_U + 16];    // row i0+2+h
            A.x  = pA[0];                   // x[i0+h]   (LDS broadcast)
            A.y  = pA[2];                   // x[i0+2+h]

            accL = __builtin_amdgcn_wmma_f32_16x16x4_f32(
                       false, A, false, BL, (short)0, accL, false, false);
            accR = __builtin_amdgcn_wmma_f32_16x16x4_f32(
                       false, A, false, BR, (short)0, accR, false, false);

            pW += 4 * THREE_U;
            pA += 4;
        }
        // All D rows are identical (A replicated over M); element 0 of lane n
        // holds column n of the fragment. Lanes 16-31 pick the right fragment
        // so lane l ends up owning global column ut*32 + l.
        res[p] = (lane < 16) ? accL[0] : accR[0];
    }

    const float* ib = Wb + NUM_K + NUM_RK;  // input bias   [1536]
    const float* rb = ib + THREE_U;         // recurrent bias [1536]
    const int u = ut * 32 + lane;

    float xz = res[0] + ib[u];
    float xr = res[1] + ib[UNITS + u];
    float xh = res[2] + ib[2 * UNITS + u];
    float rz = res[3] + rb[u];
    float rr = res[4] + rb[UNITS + u];
    float rh = res[5] + rb[2 * UNITS + u];

    float z  = hsig(xz + rz);
    float r  = hsig(xr + rr);
    float th = tanhf(xh + r * rh);
    float hp = sh[u];
    out[(size_t)b * UNITS + u] = z * hp + (1.0f - z) * th;
}

extern "C" void kernel_launch(void* const* d_in, const int* in_sizes, int n_in,
                              void* d_out, int out_size, void* d_ws, size_t ws_size,
                              hipStream_t stream) {
    const float* inputs = (const float*)d_in[0];
    const float* params = (const float*)d_in[1];
    const float* h      = (const float*)d_in[2];
    float* out = (float*)d_out;

    dim3 grid(BATCH * (UNITS / 32));  // 4096 single-wave workgroups
    dim3 block(32);
    hypergru_wmma<<<grid, block, 0, stream>>>(inputs, params, h, out);
}